// Combine_58480274702681
// MI455X (gfx1250) — compile-verified
//
#include <hip/hip_runtime.h>
#include <cstdint>

// Problem constants from the reference: B=4096 rows, D=4096 features, N=4 branches.
#define D_DIM           4096
#define ROW_BYTES       (D_DIM * 4)            // 16 KiB per row (fp32)
#define WAVES_PER_BLOCK 4                      // 128 threads/block, 64 KiB LDS/block
#define CHUNK_BYTES     512                    // wave32 * 16 B per async b128 op
#define CHUNKS          (ROW_BYTES / CHUNK_BYTES)  // 32 chunks per row

__global__ __launch_bounds__(WAVES_PER_BLOCK * 32)
void moe_select_copy(const float* __restrict__ b0,
                     const float* __restrict__ b1,
                     const float* __restrict__ b2,
                     const float* __restrict__ b3,
                     const float* __restrict__ gate,
                     float* __restrict__ out,
                     int rows)
{
    // Per-wave 16 KiB staging buffer for the async global->LDS->global stream.
    __shared__ char ldsbuf[WAVES_PER_BLOCK * ROW_BYTES];

    const int lane = threadIdx.x & 31;
    const int wave = threadIdx.x >> 5;
    const int row  = blockIdx.x * WAVES_PER_BLOCK + wave;
    if (row >= rows) return;   // wave-uniform branch

    // argmax over the 4 gate logits (strict '>' keeps first-max, matching jnp.argmax)
    const float4 g = reinterpret_cast<const float4*>(gate)[row];
    const float* src = b0;
    float best = g.x;
    if (g.y > best) { best = g.y; src = b1; }
    if (g.z > best) { best = g.z; src = b2; }
    if (g.w > best) { best = g.w; src = b3; }

    // Per-lane base addresses: lane covers bytes [lane*16, lane*16+16) of each 512B chunk.
    const unsigned long long gsrc =
        (unsigned long long)(src + (size_t)row * D_DIM) + (unsigned)(lane * 16);
    const unsigned long long gdst =
        (unsigned long long)(out + (size_t)row * D_DIM) + (unsigned)(lane * 16);
    // Flat->LDS: low 32 bits of the flat shared address are the workgroup-relative
    // LDS byte offset, which is exactly what the async ops' VDST/VSRC expects.
    const unsigned lbase =
        (unsigned)(uintptr_t)(&ldsbuf[wave * ROW_BYTES + lane * 16]);

    // Burst: 32 async global->LDS b128 transfers. The IOFFSET immediate is added to
    // BOTH the LDS and global addresses by hardware, so base VGPRs are reused.
    #pragma unroll
    for (int c = 0; c < CHUNKS; ++c) {
        asm volatile("global_load_async_to_lds_b128 %0, %1, off offset:%2"
                     :
                     : "v"(lbase), "v"(gsrc), "i"(c * CHUNK_BYTES)
                     : "memory");
    }

    // Drain the async queue before reading the staged row back out of LDS.
    asm volatile("s_wait_asynccnt 0x0" ::: "memory");

    // Burst: 32 async LDS->global b128 transfers.
    #pragma unroll
    for (int c = 0; c < CHUNKS; ++c) {
        asm volatile("global_store_async_from_lds_b128 %0, %1, off offset:%2"
                     :
                     : "v"(gdst), "v"(lbase), "i"(c * CHUNK_BYTES)
                     : "memory");
    }

    // Ensure all async stores retired (S_ENDPGM also implies wait-idle; explicit for clarity).
    asm volatile("s_wait_asynccnt 0x0" ::: "memory");
}

extern "C" void kernel_launch(void* const* d_in, const int* in_sizes, int n_in,
                              void* d_out, int out_size, void* d_ws, size_t ws_size,
                              hipStream_t stream)
{
    const float* b0   = (const float*)d_in[0];
    const float* b1   = (const float*)d_in[1];
    const float* b2   = (const float*)d_in[2];
    const float* b3   = (const float*)d_in[3];
    const float* gate = (const float*)d_in[4];
    float*       out  = (float*)d_out;

    const int rows   = in_sizes[4] / 4;                       // B = 4096
    const int blocks = (rows + WAVES_PER_BLOCK - 1) / WAVES_PER_BLOCK;

    moe_select_copy<<<blocks, WAVES_PER_BLOCK * 32, 0, stream>>>(
        b0, b1, b2, b3, gate, out, rows);
}